// LWTA_38929583571343
// MI455X (gfx1250) — compile-verified
//
#include <hip/hip_runtime.h>

// LWTA with POOL_SIZE=4 over f32 (4096x8192). Pure HBM-streaming kernel:
// 256 MiB total traffic -> ~11.5 us floor at 23.3 TB/s. One lane == one
// pool group == one 128-bit load/store, non-temporal (stream > 192MB L2,
// zero reuse). Fast path: 4 groups/thread, branch-free, constant strides so
// all 4 loads + 4 stores share one base address with immediate offsets.

typedef __attribute__((ext_vector_type(4))) float v4f;

static __device__ __forceinline__ v4f lwta4(v4f v) {
    // Tournament with first-max tie semantics (matches jnp.argmax):
    // later element must be strictly greater to beat an earlier one.
    const bool c01 = v.y > v.x;              // pair 0: x vs y (tie -> x)
    const bool c23 = v.w > v.z;              // pair 1: z vs w (tie -> z)
    const float w01 = c01 ? v.y : v.x;
    const float w23 = c23 ? v.w : v.z;
    const bool c = w23 > w01;                // final: tie -> earlier pair
    v4f o;
    o.x = (!c && !c01) ? v.x : 0.0f;
    o.y = (!c &&  c01) ? v.y : 0.0f;
    o.z = ( c && !c23) ? v.z : 0.0f;
    o.w = ( c &&  c23) ? v.w : 0.0f;
    return o;
}

#define LWTA_BLOCK 256u

// Branch-free fast path: grid * LWTA_BLOCK * 4 == n_groups (checked on host).
__global__ __launch_bounds__(LWTA_BLOCK) void lwta_pool4_x4_kernel(
    const v4f* __restrict__ in, v4f* __restrict__ out) {
    const unsigned base = blockIdx.x * (LWTA_BLOCK * 4u) + threadIdx.x;
    const v4f* __restrict__ pi = in + base;
    v4f* __restrict__ po = out + base;

    // Constant strides -> one base address, immediate offsets, 4 B128 loads
    // in flight per lane before any compute/store.
    v4f a0 = __builtin_nontemporal_load(pi);
    v4f a1 = __builtin_nontemporal_load(pi + LWTA_BLOCK);
    v4f a2 = __builtin_nontemporal_load(pi + 2u * LWTA_BLOCK);
    v4f a3 = __builtin_nontemporal_load(pi + 3u * LWTA_BLOCK);

    __builtin_nontemporal_store(lwta4(a0), po);
    __builtin_nontemporal_store(lwta4(a1), po + LWTA_BLOCK);
    __builtin_nontemporal_store(lwta4(a2), po + 2u * LWTA_BLOCK);
    __builtin_nontemporal_store(lwta4(a3), po + 3u * LWTA_BLOCK);
}

// Generic fallback: one group per thread, single bounds check.
__global__ __launch_bounds__(LWTA_BLOCK) void lwta_pool4_kernel(
    const v4f* __restrict__ in, v4f* __restrict__ out, unsigned n_groups) {
    const unsigned i = blockIdx.x * LWTA_BLOCK + threadIdx.x;
    if (i < n_groups) {
        v4f a = __builtin_nontemporal_load(in + i);
        __builtin_nontemporal_store(lwta4(a), out + i);
    }
}

extern "C" void kernel_launch(void* const* d_in, const int* in_sizes, int n_in,
                              void* d_out, int out_size, void* d_ws, size_t ws_size,
                              hipStream_t stream) {
    (void)n_in; (void)out_size; (void)d_ws; (void)ws_size;
    const v4f* in = (const v4f*)d_in[0];
    v4f* out = (v4f*)d_out;

    const unsigned n_elems  = (unsigned)in_sizes[0];   // 33,554,432
    const unsigned n_groups = n_elems / 4u;            // 8,388,608 (2^23)

    const unsigned groups_per_block = LWTA_BLOCK * 4u; // 1024

    if (n_groups % groups_per_block == 0u) {
        const unsigned grid = n_groups / groups_per_block;  // 8192 blocks
        lwta_pool4_x4_kernel<<<grid, LWTA_BLOCK, 0, stream>>>(in, out);
    } else {
        const unsigned grid = (n_groups + LWTA_BLOCK - 1u) / LWTA_BLOCK;
        lwta_pool4_kernel<<<grid, LWTA_BLOCK, 0, stream>>>(in, out, n_groups);
    }
}